// SGATLayer_28235115003922
// MI455X (gfx1250) — compile-verified
//
#include <hip/hip_runtime.h>
#include <math.h>

// ---------------------------------------------------------------------------
// SGAT layer for MI455X (gfx1250, wave32).
//   z = h @ W_fc                         -> WMMA f32 16x16x4 GEMM
//   a_src/a_dst = z @ Wa_{src,dst}       -> wave-per-node reduction
//   e = leaky(a_src[src]+a_dst[dst]+embed*cfeat), segment softmax over dst,
//   out = segment_sum(alpha * z[src])    -> wave-per-edge atomic scatter
// ---------------------------------------------------------------------------

#define N_NODES 50000
#define N_EDGES 800000
#define IN_DIM  128
#define OUT_DIM 64
#define NEG_SLOPE 0.01f

typedef __attribute__((ext_vector_type(2))) float v2f;
typedef __attribute__((ext_vector_type(8))) float v8f;

// ---------------- workspace layout (floats) --------------------------------
// [0]                 z        : N_NODES*OUT_DIM
// [Z_OFF+...]         a_src    : N_NODES
//                     a_dst    : N_NODES
//                     seg_max  : N_NODES
//                     denom    : N_NODES
//                     e/ex     : N_EDGES   (reused in place)
//                     cfeat    : 1
#define Z_OFF      0
#define ASRC_OFF   (Z_OFF + (size_t)N_NODES * OUT_DIM)
#define ADST_OFF   (ASRC_OFF + N_NODES)
#define SMAX_OFF   (ADST_OFF + N_NODES)
#define DEN_OFF    (SMAX_OFF + N_NODES)
#define E_OFF      (DEN_OFF + N_NODES)
#define CFEAT_OFF  (E_OFF + N_EDGES)

// ---------------------------------------------------------------------------
// Kernel 1: init out/denom/seg_max; compute cfeat = dot(W_feat, Wa_feat).
// ---------------------------------------------------------------------------
__global__ void init_kernel(float* __restrict__ out,
                            float* __restrict__ seg_max,
                            float* __restrict__ denom,
                            float* __restrict__ cfeat,
                            const float* __restrict__ W_feat,
                            const float* __restrict__ W_attn) {
    int tid = blockIdx.x * blockDim.x + threadIdx.x;
    if (tid < N_NODES * OUT_DIM) out[tid] = 0.0f;
    if (tid < N_NODES) {
        seg_max[tid] = -INFINITY;
        denom[tid]   = 0.0f;
    }
    if (tid == 0) {
        float c = 0.0f;
        #pragma unroll
        for (int j = 0; j < OUT_DIM; ++j)
            c += W_feat[j] * W_attn[2 * OUT_DIM + j];   // Wa_feat = W_attn[128:192]
        cfeat[0] = c;
    }
}

// ---------------------------------------------------------------------------
// Kernel 2: z = h @ W_fc via V_WMMA_F32_16X16X4_F32.
// Block = 128 threads = 4 waves; wave w owns N-tile [16w,16w+16).
// Grid.x = 3125 M-tiles (50000/16 exact). K = 128 in steps of 4.
//
// A frag (16x4 f32, 2 VGPRs/lane): lanes 0-15 hold K={k,k+1} of row M=lane,
// lanes 16-31 hold K={k+2,k+3} of row M=lane-16 (ISA 7.12.2).
// B frag mirrors the K split across lane halves; one row of B per VGPR half.
// C/D (16x16 f32, 8 VGPRs): acc[r] = C[M = 8*half + r][N = lane&15].
// ---------------------------------------------------------------------------
__global__ __launch_bounds__(128) void gemm_z_kernel(const float* __restrict__ h,
                                                     const float* __restrict__ Wfc,
                                                     float* __restrict__ z) {
    const int lane  = threadIdx.x & 31;
    const int wave  = threadIdx.x >> 5;       // 0..3 -> N tile
    const int half  = lane >> 4;              // 0: K lo pair, 1: K hi pair
    const int l16   = lane & 15;
    const int mBase = blockIdx.x * 16;
    const int nBase = wave * 16;

    const float* __restrict__ hRow = h + (size_t)(mBase + l16) * IN_DIM;
    __builtin_prefetch(hRow, 0, 1);           // global_prefetch_b8

    v8f acc = {};
    #pragma unroll
    for (int k = 0; k < IN_DIM; k += 4) {
        const int ka = k + half * 2;
        v2f a, b;
        a.x = hRow[ka + 0];
        a.y = hRow[ka + 1];
        b.x = Wfc[(ka + 0) * OUT_DIM + nBase + l16];
        b.y = Wfc[(ka + 1) * OUT_DIM + nBase + l16];
        // 8 args: (neg_a, A, neg_b, B, c_mod, C, reuse_a, reuse_b)
        acc = __builtin_amdgcn_wmma_f32_16x16x4_f32(
            false, a, false, b, (short)0, acc, false, false);
    }

    #pragma unroll
    for (int r = 0; r < 8; ++r) {
        const int orow = mBase + half * 8 + r;
        z[(size_t)orow * OUT_DIM + nBase + l16] = acc[r];
    }
}

// ---------------------------------------------------------------------------
// Kernel 3: a_src[n] = z[n,:]·Wa_src, a_dst[n] = z[n,:]·Wa_dst.
// One wave per node; lane covers dims {lane, lane+32}; shuffle reduction.
// ---------------------------------------------------------------------------
__global__ void node_attn_kernel(const float* __restrict__ z,
                                 const float* __restrict__ W_attn,
                                 float* __restrict__ a_src,
                                 float* __restrict__ a_dst) {
    const int node = blockIdx.x * (blockDim.x >> 5) + (threadIdx.x >> 5);
    const int lane = threadIdx.x & 31;
    if (node >= N_NODES) return;
    const float* __restrict__ zr = z + (size_t)node * OUT_DIM;
    const float z0 = zr[lane], z1 = zr[lane + 32];
    float s = z0 * W_attn[lane]           + z1 * W_attn[lane + 32];
    float d = z0 * W_attn[OUT_DIM + lane] + z1 * W_attn[OUT_DIM + lane + 32];
    #pragma unroll
    for (int off = 16; off > 0; off >>= 1) {
        s += __shfl_down(s, off, 32);
        d += __shfl_down(d, off, 32);
    }
    if (lane == 0) { a_src[node] = s; a_dst[node] = d; }
}

// ---------------------------------------------------------------------------
// Kernel 4: edge scores + atomic segment max (bit-trick float max).
// ---------------------------------------------------------------------------
__device__ __forceinline__ void atomicMaxF(float* addr, float val) {
    if (val >= 0.0f)
        atomicMax((int*)addr, __float_as_int(val));
    else
        atomicMin((unsigned int*)addr, (unsigned int)__float_as_int(val));
}

__global__ void edge_score_kernel(const float* __restrict__ a_src,
                                  const float* __restrict__ a_dst,
                                  const float* __restrict__ embed,
                                  const int* __restrict__ src,
                                  const int* __restrict__ dst,
                                  const float* __restrict__ cfeat,
                                  float* __restrict__ e,
                                  float* __restrict__ seg_max) {
    const int i = blockIdx.x * blockDim.x + threadIdx.x;
    if (i >= N_EDGES) return;
    const int d = dst[i];
    float v = a_src[src[i]] + a_dst[d] + embed[i] * cfeat[0];
    v = (v >= 0.0f) ? v : NEG_SLOPE * v;
    e[i] = v;
    atomicMaxF(&seg_max[d], v);
}

// ---------------------------------------------------------------------------
// Kernel 5: ex = exp(e - seg_max[dst]); denom[dst] += ex (ex overwrites e).
// ---------------------------------------------------------------------------
__global__ void edge_exp_kernel(const int* __restrict__ dst,
                                const float* __restrict__ seg_max,
                                float* __restrict__ e,
                                float* __restrict__ denom) {
    const int i = blockIdx.x * blockDim.x + threadIdx.x;
    if (i >= N_EDGES) return;
    const int d = dst[i];
    const float v = __expf(e[i] - seg_max[d]);
    e[i] = v;
    atomicAdd(&denom[d], v);
}

// ---------------------------------------------------------------------------
// Kernel 6: out[dst,:] += (ex/denom[dst]) * z[src,:].
// One wave per edge; lane handles dims {lane, lane+32}.
// ---------------------------------------------------------------------------
__global__ void scatter_kernel(const float* __restrict__ z,
                               const float* __restrict__ ex,
                               const float* __restrict__ denom,
                               const int* __restrict__ src,
                               const int* __restrict__ dst,
                               float* __restrict__ out) {
    const int wid  = blockIdx.x * (blockDim.x >> 5) + (threadIdx.x >> 5);
    const int lane = threadIdx.x & 31;
    if (wid >= N_EDGES) return;
    const int s = src[wid], d = dst[wid];
    const float alpha = ex[wid] / denom[d];
    const float* __restrict__ zr = z + (size_t)s * OUT_DIM;
    float* __restrict__ orow = out + (size_t)d * OUT_DIM;
    atomicAdd(&orow[lane],      alpha * zr[lane]);
    atomicAdd(&orow[lane + 32], alpha * zr[lane + 32]);
}

// ---------------------------------------------------------------------------
extern "C" void kernel_launch(void* const* d_in, const int* in_sizes, int n_in,
                              void* d_out, int out_size, void* d_ws, size_t ws_size,
                              hipStream_t stream) {
    const float* h      = (const float*)d_in[0];
    const float* embed  = (const float*)d_in[1];
    const int*   src    = (const int*)d_in[2];
    const int*   dst    = (const int*)d_in[3];
    const float* W_fc   = (const float*)d_in[4];
    const float* W_attn = (const float*)d_in[5];
    const float* W_feat = (const float*)d_in[6];
    float* out = (float*)d_out;

    float* ws      = (float*)d_ws;
    float* z       = ws + Z_OFF;
    float* a_src   = ws + ASRC_OFF;
    float* a_dst   = ws + ADST_OFF;
    float* seg_max = ws + SMAX_OFF;
    float* denom   = ws + DEN_OFF;
    float* e       = ws + E_OFF;
    float* cfeat   = ws + CFEAT_OFF;

    // 1. init out / seg_max / denom / cfeat
    {
        int n = N_NODES * OUT_DIM;
        init_kernel<<<(n + 255) / 256, 256, 0, stream>>>(out, seg_max, denom,
                                                         cfeat, W_feat, W_attn);
    }
    // 2. z = h @ W_fc  (WMMA)  — 3125 M-tiles, 4 waves cover N=64
    gemm_z_kernel<<<N_NODES / 16, 128, 0, stream>>>(h, W_fc, z);
    // 3. per-node attention terms — 8 waves/block
    node_attn_kernel<<<(N_NODES + 7) / 8, 256, 0, stream>>>(z, W_attn, a_src, a_dst);
    // 4. edge scores + segment max
    edge_score_kernel<<<(N_EDGES + 255) / 256, 256, 0, stream>>>(a_src, a_dst, embed,
                                                                 src, dst, cfeat,
                                                                 e, seg_max);
    // 5. exp + denom
    edge_exp_kernel<<<(N_EDGES + 255) / 256, 256, 0, stream>>>(dst, seg_max, e, denom);
    // 6. weighted scatter-sum — one wave per edge
    scatter_kernel<<<(N_EDGES + 7) / 8, 256, 0, stream>>>(z, e, denom, src, dst, out);
}